// SoftToHardNdEncoder_65609920414450
// MI455X (gfx1250) — compile-verified
//
#include <hip/hip_runtime.h>
#include <hip/hip_bf16.h>

// CDNA5 wave32. Matrix ops: V_WMMA_F32_16X16X4_F32. Staging: TENSOR_LOAD_TO_LDS (TDM).
typedef __attribute__((ext_vector_type(2))) float v2f;          // f32 A/B operand (16x4 / 4x16)
typedef __attribute__((ext_vector_type(8))) float v8f;          // f32 C/D operand (16x16)
typedef __attribute__((ext_vector_type(4))) unsigned int u32x4; // TDM D# group 0
typedef __attribute__((ext_vector_type(4))) int i32x4;          // TDM D# groups 2/3
typedef __attribute__((ext_vector_type(8))) int i32x8;          // TDM D# group 1

#if defined(__has_builtin)
#if __has_builtin(__builtin_amdgcn_tensor_load_to_lds) && \
    __has_builtin(__builtin_amdgcn_s_wait_tensorcnt)
#define USE_TDM 1
#endif
#endif
#ifndef USE_TDM
#define USE_TDM 0
#endif

#if __has_builtin(__builtin_amdgcn_exp2f)
#define FAST_EXP2(x) __builtin_amdgcn_exp2f(x)   // raw v_exp_f32
#else
#define FAST_EXP2(x) exp2f(x)
#endif
#if __has_builtin(__builtin_amdgcn_sqrtf)
#define FAST_SQRT(x) __builtin_amdgcn_sqrtf(x)   // raw v_sqrt_f32, no OCML fixup
#else
#define FAST_SQRT(x) sqrtf(x)
#endif

namespace {
constexpr int kL = 8;        // latent dims
constexpr int kK = 512;      // codes per latent
constexpr int kD = 16;       // channel dim per latent
constexpr int kC = 128;      // total channels
constexpr int kHW = 1024;    // 32*32
constexpr int kPos = 16384;  // B*H*W
constexpr int kTiles = kK / 16;
constexpr float kLog2e = 1.4426950408889634f;
}

// xor-16 lane swap within wave32 via ds_swizzle (SWAPX16: xor=0x10, and=0x1f)
__device__ __forceinline__ float swap16f(float x) {
  return __int_as_float(__builtin_amdgcn_ds_swizzle(__float_as_int(x), 0x401f));
}
__device__ __forceinline__ int swap16i(int x) {
  return __builtin_amdgcn_ds_swizzle(x, 0x401f);
}
// Code-row permutation: GEMM1 A-tile row r holds actual (tile-local) code sfwd(r).
// Chosen so GEMM1's C-layout weight registers are exactly GEMM2's B-layout operands.
__device__ __forceinline__ int sfwd(int r) { return ((r & 6) << 1) | ((r >> 3) << 1) | (r & 1); }
__device__ __forceinline__ int sinv(int x) { return ((x & 2) << 2) | ((x >> 2) << 1) | (x & 1); }

// One wave: 16 consecutive positions x one latent l x all 512 codes (flash-style online softmin).
// Workgroup: 8 waves, same l, 128 consecutive positions; codes[l] staged once into LDS.
__global__ __launch_bounds__(256) void vq_softhard_kernel(
    const float* __restrict__ z, const float* __restrict__ codes,
    float* __restrict__ soft, float* __restrict__ hard, int* __restrict__ idxes) {
  __shared__ float lds_codes[kK * kD];  // 32 KB, LINEAR row-major [code][d]
  __shared__ float lds_w2[kK];          // 2 KB, ||code||^2 stored PERMUTED (sinv)

  const int l = blockIdx.x & 7;
  const int posBase = (blockIdx.x >> 3) * 128;
  const int tid = threadIdx.x;

#if USE_TDM
  // ---- Tensor Data Mover: DMA codes[l] (16 x 512 x 4B tile) straight into LDS ----
  if (tid < 32) {  // one wave issues; TDM ignores EXEC, tracked by TENSORcnt
    unsigned long long ga =
        (unsigned long long)(const void*)(codes + (size_t)l * kK * kD);
    unsigned int laddr = (unsigned int)(uintptr_t)(&lds_codes[0]);  // LDS byte offset
    u32x4 g0;
    g0[0] = 1u;                                                  // count=1, no gather
    g0[1] = laddr;                                               // lds_addr [63:32]
    g0[2] = (unsigned int)(ga & 0xffffffffu);                    // global_addr [95:64]
    g0[3] = (unsigned int)((ga >> 32) & 0x1ffffffu) | (2u << 30);// addr hi + type=2
    i32x8 g1;
    g1[0] = 0x00020000;          // workgroup_mask=0, data_size=2 (4 bytes)
    g1[1] = (int)(16u << 16);    // tensor_dim0 = 16 (elements per row)
    g1[2] = (int)(512u << 16);   // tensor_dim1 = 512 rows
    g1[3] = (int)(16u << 16);    // tile_dim0 = 16
    g1[4] = 512;                 // tile_dim1 = 512, tile_dim2 = 0
    g1[5] = 16;                  // tensor_dim0_stride = 16
    g1[6] = 0;
    g1[7] = 0;                   // tensor_dim1_stride = 0 (unused, 2D tile)
    i32x4 z4 = {};
#if __clang_major__ >= 23
    i32x8 z8 = {};
    __builtin_amdgcn_tensor_load_to_lds(g0, g1, z4, z4, z8, 0);
#else
    __builtin_amdgcn_tensor_load_to_lds(g0, g1, z4, z4, 0);
#endif
    __builtin_amdgcn_s_wait_tensorcnt(0);
  }
  __syncthreads();
  // ||w||^2 from LDS, stored at permuted slot so the hot loop reads it contiguously
  for (int c = tid; c < kK; c += 256) {
    const float4* row = reinterpret_cast<const float4*>(lds_codes + c * kD);
    float4 q0 = row[0], q1 = row[1], q2 = row[2], q3 = row[3];
    float w2 = q0.x*q0.x + q0.y*q0.y + q0.z*q0.z + q0.w*q0.w
             + q1.x*q1.x + q1.y*q1.y + q1.z*q1.z + q1.w*q1.w
             + q2.x*q2.x + q2.y*q2.y + q2.z*q2.z + q2.w*q2.w
             + q3.x*q3.x + q3.y*q3.y + q3.z*q3.z + q3.w*q3.w;
    lds_w2[(c & ~15) | sinv(c & 15)] = w2;
  }
  __syncthreads();
#else
  // ---- fallback staging: coalesced float4 global->LDS, w2 on the fly ----
  {
    const float4* src = reinterpret_cast<const float4*>(codes + (size_t)l * kK * kD);
    float4* dst = reinterpret_cast<float4*>(lds_codes);
    for (int c = tid; c < kK; c += 256) {
      float4 q0 = src[c*4+0], q1 = src[c*4+1], q2 = src[c*4+2], q3 = src[c*4+3];
      dst[c*4+0] = q0; dst[c*4+1] = q1; dst[c*4+2] = q2; dst[c*4+3] = q3;
      float w2 = q0.x*q0.x + q0.y*q0.y + q0.z*q0.z + q0.w*q0.w
               + q1.x*q1.x + q1.y*q1.y + q1.z*q1.z + q1.w*q1.w
               + q2.x*q2.x + q2.y*q2.y + q2.z*q2.z + q2.w*q2.w
               + q3.x*q3.x + q3.y*q3.y + q3.z*q3.z + q3.w*q3.w;
      lds_w2[(c & ~15) | sinv(c & 15)] = w2;
    }
  }
  __syncthreads();
#endif

  const int wave = tid >> 5;
  const int lane = tid & 31;
  const int laneLo = lane & 15;
  const int hi = lane >> 4;
  const int rowA = sfwd(laneLo);  // permuted GEMM1 A row for this lane

  const int pos = posBase + wave * 16 + laneLo;  // tiles never cross batch (1024%16==0)
  const int b = pos >> 10;
  const int hw = pos & 1023;
  const float* zb = z + (size_t)b * kC * kHW + (size_t)(l * kD) * kHW + hw;

  // GEMM1 B operand (h tile), loop invariant: vgpr j -> K = kb*4 + 2*hi + j, N = laneLo.
  v2f hB[4];
#pragma unroll
  for (int kb = 0; kb < 4; ++kb) {
    hB[kb][0] = zb[(size_t)(kb * 4 + 2 * hi + 0) * kHW];
    hB[kb][1] = zb[(size_t)(kb * 4 + 2 * hi + 1) * kHW];
  }
  // ||h||^2: this lane holds 8 of its position's dims, xor-16 partner the other 8.
  float h2p = 0.f;
#pragma unroll
  for (int kb = 0; kb < 4; ++kb) h2p += hB[kb][0]*hB[kb][0] + hB[kb][1]*hB[kb][1];
  h2p += swap16f(h2p);

  // Online softmin state. All distances kept pre-scaled by log2(e) so weights are
  // a single v_exp_f32 (base-2); softmax is invariant to this uniform scaling.
  float m_run = 3.0e38f;
  float S = 0.f;
  v8f acc = {};           // GEMM2 C accumulator: out^T tile (M=d rows, N=pos lanes)
  float bestD = 3.0e38f;
  int bestI = 0;

  for (int ct = 0; ct < kTiles; ++ct) {
    // ---- GEMM1: cross^T = codes(16c x 16d) * h(16d x 16p), 4 x WMMA K=4 ----
    v8f cr = {};
    const float* aBase = lds_codes + (ct * 16 + rowA) * kD;
#pragma unroll
    for (int kb = 0; kb < 4; ++kb) {
      v2f a; a[0] = aBase[kb * 4 + 2 * hi]; a[1] = aBase[kb * 4 + 2 * hi + 1];
      cr = __builtin_amdgcn_wmma_f32_16x16x4_f32(false, a, false, hB[kb],
                                                 (short)0, cr, false, false);
    }
    // ---- distances + argmin + tile min (C row q <-> local code ((q&6)<<1)|(hi<<1)|(q&1)) ----
    const float* wp = lds_w2 + ct * 16 + 8 * hi;  // permuted store -> contiguous read
    float dist[8];
    float tmin = 3.0e38f;
#pragma unroll
    for (int q = 0; q < 8; ++q) {
      float d2 = fmaxf(h2p - 2.f * cr[q] + wp[q], 1e-12f);
      float dd = FAST_SQRT(d2) * kLog2e;          // scaled distance
      dist[q] = dd;
      int ci = ct * 16 + (((q & 6) << 1) | (hi << 1) | (q & 1));  // ascending in q
      bool better = dd < bestD;                   // strict < keeps first (lowest) index
      bestD = better ? dd : bestD;
      bestI = better ? ci : bestI;
      tmin = fminf(tmin, dd);
    }
    tmin = fminf(tmin, swap16f(tmin));            // shared reference across lane pair
    float m_new = fminf(m_run, tmin);
    float f = FAST_EXP2(m_new - m_run);           // rescale <= 1 (0 on first tile)
    m_run = m_new;
    S *= f;
#pragma unroll
    for (int j = 0; j < 8; ++j) acc[j] *= f;      // per-position accumulator rescale
    float w[8];
#pragma unroll
    for (int q = 0; q < 8; ++q) { w[q] = FAST_EXP2(m_new - dist[q]); S += w[q]; }

    // ---- GEMM2: out^T += codes^T(16d x 16c) * probs(16c x 16p), 4 x WMMA K=4 ----
    // Permutation makes B operands pure register renames of w[] (no shuffles).
#pragma unroll
    for (int kb = 0; kb < 4; ++kb) {
      v2f b2; b2[0] = w[2 * kb]; b2[1] = w[2 * kb + 1];
      int abase = (ct * 16 + kb * 4 + 2 * hi) * kD + laneLo;  // linear code storage
      v2f a2; a2[0] = lds_codes[abase]; a2[1] = lds_codes[abase + kD];
      acc = __builtin_amdgcn_wmma_f32_16x16x4_f32(false, a2, false, b2,
                                                  (short)0, acc, false, false);
    }
  }

  // ---- merge lane-pair state (symmetric -> both lanes agree) ----
  S += swap16f(S);
  {
    float oD = swap16f(bestD);
    int oI = swap16i(bestI);
    bool take = (oD < bestD) || (oD == bestD && oI < bestI);
    bestD = take ? oD : bestD;
    bestI = take ? oI : bestI;
  }
  float inv = 1.f / S;

  // out^T C layout: lane-lo -> d = j, lane-hi -> d = j+8; N = laneLo = position
  size_t obase = (size_t)pos * kC + l * kD + hi * 8;  // 32B aligned
  float4 s0 = make_float4(acc[0]*inv, acc[1]*inv, acc[2]*inv, acc[3]*inv);
  float4 s1 = make_float4(acc[4]*inv, acc[5]*inv, acc[6]*inv, acc[7]*inv);
  *reinterpret_cast<float4*>(soft + obase) = s0;
  *reinterpret_cast<float4*>(soft + obase + 4) = s1;
  const float4* hp = reinterpret_cast<const float4*>(lds_codes + bestI * kD + hi * 8);
  float4 h0 = hp[0], h1 = hp[1];
  *reinterpret_cast<float4*>(hard + obase) = h0;
  *reinterpret_cast<float4*>(hard + obase + 4) = h1;
  if (hi == 0) idxes[(size_t)pos * kL + l] = bestI;  // after all WMMAs (EXEC rule)
}

extern "C" void kernel_launch(void* const* d_in, const int* in_sizes, int n_in,
                              void* d_out, int out_size, void* d_ws, size_t ws_size,
                              hipStream_t stream) {
  const float* z = (const float*)d_in[0];      // [16,128,32,32] f32
  const float* codes = (const float*)d_in[1];  // [8,512,16] f32
  // d_out = soft [16384*128] f32 | hard [16384*128] f32 | idxes [16384*8] i32
  float* soft = (float*)d_out;
  float* hard = soft + (size_t)kPos * kC;
  int* idxes = (int*)(hard + (size_t)kPos * kC);
  dim3 grid((kPos / 128) * kL);  // 1024 workgroups: (pos-block, latent)
  dim3 block(256);               // 8 wave32s
  vq_softhard_kernel<<<grid, block, 0, stream>>>(z, codes, soft, hard, idxes);
  (void)in_sizes; (void)n_in; (void)out_size; (void)d_ws; (void)ws_size;
}